// CutOut_59201829208648
// MI455X (gfx1250) — compile-verified
//
#include <hip/hip_runtime.h>
#include <hip/hip_bf16.h>

// CutOut: images [B=64, H=512, W=512, C=3] f32, zero a length x length square
// centered at (center_y[b], center_x[b]) per sample. Pure streaming op:
// ~402 MB of traffic, ~17us floor at 23.3 TB/s. Optimized as a coalesced
// 128-bit non-temporal copy with a block-uniform row test.

typedef float v4f __attribute__((ext_vector_type(4)));

#define IMG_B 64
#define IMG_H 512
#define IMG_W 512
#define IMG_C 3
#define ROW_FLOATS (IMG_W * IMG_C)   // 1536 floats per row
#define ROW_VEC4   (ROW_FLOATS / 4)  // 384 float4 per row

__global__ __launch_bounds__(ROW_VEC4) void CutOut_kernel(
    const float* __restrict__ img,
    const int*   __restrict__ center_y,
    const int*   __restrict__ center_x,
    const int*   __restrict__ length_p,
    float*       __restrict__ out)
{
    // One block per (b, h) row. b and h are block-uniform -> scalar branches,
    // scalar loads of centers.
    const int row = blockIdx.x;          // 0 .. B*H-1
    const int b   = row >> 9;            // row / 512
    const int h   = row & (IMG_H - 1);   // row % 512
    const int t   = threadIdx.x;         // 0 .. 383

    const long long base = (long long)row * ROW_VEC4 + t;  // float4 index

    // Non-temporal 128-bit load: stream, don't pollute L2 (192MB < 402MB stream).
    v4f v = __builtin_nontemporal_load(((const v4f*)img) + base);

    const int half = length_p[0] >> 1;   // length/2 (device scalar)
    const int cy   = center_y[b];
    const int y0   = cy - half;
    const int y1   = cy + half;

    // Block-uniform row test: ~90% of blocks skip all per-element math.
    if (h >= y0 && h < y1) {
        const int cx = center_x[b];
        const int x0 = cx - half;
        const int x1 = cx + half;
        // Elements e = t*4+k within the row; pixel column = e / 3.
        // Track (col, rem) incrementally to avoid 4 divides.
        const int e0  = t * 4;
        int col = e0 / 3;                // one magic-multiply divide
        int rem = e0 - col * 3;
        #pragma unroll
        for (int k = 0; k < 4; ++k) {
            if (col >= x0 && col < x1) v[k] = 0.0f;
            if (++rem == 3) { rem = 0; ++col; }
        }
    }

    __builtin_nontemporal_store(v, ((v4f*)out) + base);
}

extern "C" void kernel_launch(void* const* d_in, const int* in_sizes, int n_in,
                              void* d_out, int out_size, void* d_ws, size_t ws_size,
                              hipStream_t stream) {
    const float* img      = (const float*)d_in[0];
    const int*   center_y = (const int*)d_in[1];
    const int*   center_x = (const int*)d_in[2];
    const int*   length_p = (const int*)d_in[3];
    float*       out      = (float*)d_out;

    const int nblocks = IMG_B * IMG_H;   // 32768 blocks, one 6KB row each
    CutOut_kernel<<<nblocks, ROW_VEC4, 0, stream>>>(img, center_y, center_x,
                                                    length_p, out);
}